// SAGPoolReadout_32014686224553
// MI455X (gfx1250) — compile-verified
//
#include <hip/hip_runtime.h>
#include <hip/hip_bf16.h>
#include <math.h>

// Problem constants (match reference)
#define G_     512
#define NPER_  256
#define D_     128
#define EPER_  4096
#define K_     128
#define N_     (G_ * NPER_)     // 131072 nodes
#define E_     (G_ * EPER_)     // 2097152 edges

typedef __attribute__((ext_vector_type(2))) float v2f;
typedef __attribute__((ext_vector_type(8))) float v8f;

// ---------------------------------------------------------------------------
// Kernel 1: xw[n] = dot(x[n,:], gcn_w)   (wave-per-node, float4 coalesced)
// ---------------------------------------------------------------------------
__global__ __launch_bounds__(256) void k_xw(const float* __restrict__ x,
                                            const float* __restrict__ w,
                                            float* __restrict__ xw) {
    int wave = (blockIdx.x * blockDim.x + threadIdx.x) >> 5;
    int lane = threadIdx.x & 31;
    if (wave >= N_) return;
    const float4* xp = (const float4*)(x + (size_t)wave * D_);
    const float4* wp = (const float4*)w;
    float4 xv = xp[lane];
    float4 wv = wp[lane];
    float d = xv.x * wv.x + xv.y * wv.y + xv.z * wv.z + xv.w * wv.w;
    // wave32 butterfly reduction
    for (int off = 16; off > 0; off >>= 1)
        d += __shfl_xor(d, off, 32);
    if (lane == 0) xw[wave] = d;
}

// ---------------------------------------------------------------------------
// Kernel 2: per-graph GCN aggregation.
// Deterministic WITHOUT serial scans:
//  - degree: LDS integer atomics (exact, order-independent)
//  - message sum: 2^-40 fixed-point in u64 LDS atomics (exact integer adds,
//    order-independent -> bit-stable across replays; quantization ~1e-12)
// score[n] = (sum_{e: dst==n} xw[src]*dinv[src]) * dinv[n] + xw[n]*dinv[n]^2 + b
// ---------------------------------------------------------------------------
__global__ __launch_bounds__(256) void k_score(const int* __restrict__ ei,
                                               const float* __restrict__ xw,
                                               const float* __restrict__ gcn_b,
                                               float* __restrict__ score) {
    __shared__ int                ssrc[EPER_];
    __shared__ int                sdst[EPER_];
    __shared__ float              sxw[NPER_];
    __shared__ float              sdinv[NPER_];
    __shared__ int                sdeg[NPER_];
    __shared__ unsigned long long sacc[NPER_];

    const int g    = blockIdx.x;
    const int tid  = threadIdx.x;
    const int base = g * NPER_;
    const int* srcp = ei + (size_t)g * EPER_;               // edge_index[0]
    const int* dstp = ei + (size_t)E_ + (size_t)g * EPER_;  // edge_index[1]

    sdeg[tid] = 1;        // self loop
    sacc[tid] = 0ull;
    sxw[tid]  = xw[base + tid];
    for (int e = tid; e < EPER_; e += 256) {
        ssrc[e] = srcp[e] - base;   // local node id 0..255
        sdst[e] = dstp[e] - base;
    }
    __syncthreads();

    // degree at target via integer LDS atomics (deterministic)
    for (int e = tid; e < EPER_; e += 256)
        atomicAdd(&sdeg[sdst[e]], 1);
    __syncthreads();

    sdinv[tid] = rsqrtf((float)sdeg[tid]);
    __syncthreads();

    // fixed-point message aggregation via u64 LDS atomics (deterministic)
    const double SCALE = 1099511627776.0;  // 2^40
    for (int e = tid; e < EPER_; e += 256) {
        int s = ssrc[e];
        float v = sxw[s] * sdinv[s];
        long long q = (long long)rint((double)v * SCALE);
        atomicAdd(&sacc[sdst[e]], (unsigned long long)q);  // 2's complement add
    }
    __syncthreads();

    float dv  = sdinv[tid];
    float agg = (float)((double)(long long)sacc[tid] * (1.0 / 1099511627776.0));
    score[base + tid] = agg * dv + sxw[tid] * dv * dv + gcn_b[0];
}

// ---------------------------------------------------------------------------
// Kernel 3: per-graph top-K mask (rank by counting, jax top_k tie-break),
// then gated mean/max pooling over kept nodes -> readout[G, 2D].
// All 256 threads stream x: thread t = (dim t&127, node-half t>>7),
// partials combined in LDS.
// ---------------------------------------------------------------------------
__global__ __launch_bounds__(256) void k_topk_pool(const float* __restrict__ x,
                                                   const float* __restrict__ score,
                                                   float* __restrict__ readout) {
    __shared__ float sl[NPER_];
    __shared__ float gate[NPER_];
    __shared__ int   keep[NPER_];
    __shared__ float pacc[256];
    __shared__ float pmx[256];

    const int g    = blockIdx.x;
    const int tid  = threadIdx.x;
    const int base = g * NPER_;

    float s = score[base + tid];
    sl[tid] = s;
    __syncthreads();

    // rank_i = #{j : s_j > s_i  or (s_j == s_i and j < i)}  -> kept iff rank < K
    int r = 0;
    for (int j = 0; j < NPER_; ++j) {
        float sj = sl[j];
        r += ((sj > s) || (sj == s && j < tid)) ? 1 : 0;
    }
    keep[tid] = (r < K_) ? 1 : 0;
    gate[tid] = tanhf(s);
    __syncthreads();

    const int d    = tid & (D_ - 1);   // dim 0..127
    const int half = tid >> 7;         // node half 0/1
    const int nh   = half * 128;
    float acc = 0.0f;
    float mx  = -INFINITY;
    const float* xg = x + (size_t)(base + nh) * D_ + d;  // coalesced across 128 lanes
    for (int n = 0; n < 128; ++n) {
        if (keep[nh + n]) {
            float v = xg[(size_t)n * D_] * gate[nh + n];
            acc += v;
            mx = fmaxf(mx, v);
        }
    }
    pacc[tid] = acc;
    pmx[tid]  = mx;
    __syncthreads();

    if (tid < D_) {
        float m = (pacc[tid] + pacc[tid + 128]) * (1.0f / (float)K_);
        float M = fmaxf(pmx[tid], pmx[tid + 128]);
        readout[(size_t)g * (2 * D_) + tid]      = m;
        readout[(size_t)g * (2 * D_) + D_ + tid] = M;
    }
}

// ---------------------------------------------------------------------------
// Kernel 4: out[512,128] = readout[512,256] @ lin_w[256,128] + lin_b
// Full-f32 WMMA: V_WMMA_F32_16X16X4_F32, one wave per 16x16 output tile.
// A 16x4 layout : lane%16 = M row, (lane/16)*2 = K pair -> 2 VGPRs/lane
// B 4x16 layout : lane%16 = N col, (lane/16)*2 = K pair -> 2 VGPRs/lane
// C/D 16x16     : 8 VGPRs; vgpr r holds M = r + 8*(lane/16), N = lane%16
// ---------------------------------------------------------------------------
__global__ __launch_bounds__(128) void k_gemm_wmma(const float* __restrict__ R,
                                                   const float* __restrict__ W,
                                                   const float* __restrict__ bias,
                                                   float* __restrict__ out) {
    const int waveInBlk = threadIdx.x >> 5;
    const int lane      = threadIdx.x & 31;
    const int tile      = blockIdx.x * 4 + waveInBlk;  // 0..255
    const int tm = tile >> 3;                          // 0..31  (M tiles: 512/16)
    const int tn = tile & 7;                           // 0..7   (N tiles: 128/16)
    const int m0 = tm * 16;
    const int n0 = tn * 16;

    const int mn = lane & 15;          // M row for A, N col for B
    const int kb = (lane >> 4) << 1;   // K sub-offset: 0 or 2

    v8f c = {};
    const float* Ar = R + (size_t)(m0 + mn) * (2 * D_);  // row of readout, K=256
    const float* Bc = W + n0 + mn;                        // column of lin_w

    for (int k0 = 0; k0 < 2 * D_; k0 += 4) {
        v2f a, b;
        a.x = Ar[k0 + kb];
        a.y = Ar[k0 + kb + 1];
        b.x = Bc[(size_t)(k0 + kb) * D_];
        b.y = Bc[(size_t)(k0 + kb + 1) * D_];
        c = __builtin_amdgcn_wmma_f32_16x16x4_f32(
                /*neg_a=*/false, a, /*neg_b=*/false, b,
                /*c_mod=*/(short)0, c, /*reuse_a=*/false, /*reuse_b=*/false);
    }

    const int hi = lane >> 4;
    const float bb = bias[n0 + mn];
    #pragma unroll
    for (int rr = 0; rr < 8; ++rr) {
        int mm = rr + 8 * hi;
        out[(size_t)(m0 + mm) * D_ + n0 + mn] = c[rr] + bb;
    }
}

// ---------------------------------------------------------------------------
// Launch
// ---------------------------------------------------------------------------
extern "C" void kernel_launch(void* const* d_in, const int* in_sizes, int n_in,
                              void* d_out, int out_size, void* d_ws, size_t ws_size,
                              hipStream_t stream) {
    const float* x     = (const float*)d_in[0];   // [N, D]
    // d_in[1]: graph_indicator (unused — equal-size graphs)
    const int*   ei    = (const int*)d_in[2];     // [2, E]
    const float* gcn_w = (const float*)d_in[3];   // [D, 1]
    const float* gcn_b = (const float*)d_in[4];   // [1]
    const float* lin_w = (const float*)d_in[5];   // [2D, D]
    const float* lin_b = (const float*)d_in[6];   // [D]
    float* out = (float*)d_out;                   // [G, D]

    // workspace partition (floats): xw[N] | score[N] | readout[G*2D]
    float* ws      = (float*)d_ws;
    float* xw      = ws;
    float* score   = ws + N_;
    float* readout = ws + 2 * N_;

    // 1) scoring matvec: wave-per-node, 8 waves/block
    k_xw<<<N_ / 8, 256, 0, stream>>>(x, gcn_w, xw);

    // 2) per-graph GCN aggregation (deterministic LDS atomics)
    k_score<<<G_, 256, 0, stream>>>(ei, xw, gcn_b, score);

    // 3) top-K + gated mean/max pooling
    k_topk_pool<<<G_, 256, 0, stream>>>(x, score, readout);

    // 4) final GEMM with f32 WMMA (256 tiles, 4 waves/block -> 64 blocks)
    k_gemm_wmma<<<64, 128, 0, stream>>>(readout, lin_w, lin_b, out);
}